// MaxPlus_50062138802180
// MI455X (gfx1250) — compile-verified
//
#include <hip/hip_runtime.h>
#include <stdint.h>

// Max-plus (tropical) matmul: out[b,o] = max( max_k(x[b,k] + W[o,k]), bias[o] )
// B=1024, IN=512, OUT=512, all f32.
//
// VALU-bound (536M f32 ops vs ~5MB of HBM traffic), so: register-blocked 8x4
// per-thread tiles, LDS tiling with TDM (tensor_load_to_lds) double-buffered
// async copies, TENSORcnt-based pipelining. WMMA is not applicable (max-plus
// is not expressible as a sum-reduction matrix op).

#define B_DIM   1024
#define IN_DIM  512
#define OUT_DIM 512

#define BM 128          // batch rows per block
#define BN 64           // output cols per block
#define BK 32           // K-tile
#define TM 8            // per-thread batch rows
#define TN 4            // per-thread out cols
#define THREADS 256     // 8 waves (wave32)
#define WAVES 8
#define KTILES (IN_DIM / BK)            // 16
#define LDS_STRIDE (BK + 1)             // 33 floats/row: TDM pads 1 DWORD per 32
#define XROWS_PER_WAVE (BM / WAVES)     // 16
#define WROWS_PER_WAVE (BN / WAVES)     // 8

#if defined(__has_builtin)
#if __has_builtin(__builtin_amdgcn_tensor_load_to_lds) && \
    __has_builtin(__builtin_amdgcn_s_wait_tensorcnt) && \
    __has_builtin(__builtin_amdgcn_readfirstlane)
#define USE_TDM 1
#endif
#endif

typedef unsigned int u32x4 __attribute__((ext_vector_type(4)));
typedef int          i32x8 __attribute__((ext_vector_type(8)));
typedef int          i32x4 __attribute__((ext_vector_type(4)));

#ifdef USE_TDM
// Issue one TDM 2-D tile load: `rows` rows of BK f32 elements from a row-major
// [*, IN_DIM] f32 matrix starting at gsrc, into LDS at lds_byte_addr with a
// 1-DWORD pad inserted after every 32 DWORDs (=> 33-float LDS row stride).
// D# packing per CDNA5 ISA sec. 8.3/8.4. This toolchain (clang-23 /
// therock-10.0) exposes the 6-arg builtin:
//   (u32x4 group0, i32x8 group1, i32x4 group2, i32x4 group3, i32x8, i32 cpol)
__device__ __forceinline__ void tdm_load_2d(const float* gsrc,
                                            uint32_t lds_byte_addr,
                                            uint32_t rows) {
  const uint64_t ga = (uint64_t)(uintptr_t)gsrc;

  // ---- D# group 0 (4 DWORDs) ----
  u32x4 g0;
  g0.x = 1u;                                   // count=1, user mode, no gather
  g0.y = lds_byte_addr;                        // lds_addr[31:0]
  g0.z = (uint32_t)(ga & 0xFFFFFFFFu);         // global_addr[31:0]
  g0.w = (uint32_t)((ga >> 32) & 0x1FFFFFFu)   // global_addr[56:32]
       | (2u << 30);                           // type = 2 ("image")

  // ---- D# group 1 (8 DWORDs) ----
  const uint32_t td0   = BK;                   // tensor_dim0 == tile_dim0 (no OOB)
  const uint32_t td1   = rows;                 // tensor_dim1 == tile_dim1
  const uint32_t tile0 = BK;
  const uint32_t tile1 = rows;
  const uint64_t str0  = IN_DIM;               // dim0 stride, data_size units

  i32x8 g1;
  g1[0] = (int)((2u << 16)        // data_size = 2 -> 4 bytes
              | (1u << 20)        // pad_enable
              | (4u << 22)        // pad_interval code 4 -> every 32 DWORDs
              | (0u << 25));      // pad_amount code 0 -> 1 DWORD
  g1[1] = (int)((td0 & 0xFFFFu) << 16);                               // dim0[15:0]
  g1[2] = (int)(((td0 >> 16) & 0xFFFFu) | ((td1 & 0xFFFFu) << 16));   // dim0[31:16], dim1[15:0]
  g1[3] = (int)(((td1 >> 16) & 0xFFFFu) | ((tile0 & 0xFFFFu) << 16)); // dim1[31:16], tile0
  g1[4] = (int)((tile1 & 0xFFFFu));                                   // tile1, tile2=0
  g1[5] = (int)(uint32_t)(str0 & 0xFFFFFFFFu);                        // stride0[31:0]
  g1[6] = (int)(uint32_t)((str0 >> 32) & 0xFFFFu);                    // stride0[47:32], stride1[15:0]=0
  g1[7] = 0;                                                          // stride1[47:16]=0

  i32x4 z4 = {0, 0, 0, 0};              // groups 2/3 unused (2-D tile, tile_dim2=0)
  i32x8 z8 = {0, 0, 0, 0, 0, 0, 0, 0};  // extra operand (zero-filled, probe-verified)
  __builtin_amdgcn_tensor_load_to_lds(g0, g1, z4, z4, z8, 0);
}
#endif

__global__ __launch_bounds__(THREADS)
void maxplus_kernel(const float* __restrict__ x,      // [B, IN]
                    const float* __restrict__ w,      // [OUT, IN]
                    const float* __restrict__ bias,   // [OUT]
                    float* __restrict__ out) {        // [B, OUT]
  __shared__ float xs[2][BM * LDS_STRIDE];   // 2 x 128 x 33 f32
  __shared__ float ws[2][BN * LDS_STRIDE];   // 2 x  64 x 33 f32

  const int tid   = (int)threadIdx.x;
  const int tx    = tid & 15;                 // 16 threads over out-cols
  const int ty    = tid >> 4;                 // 16 threads over batch-rows
  const int bRow0 = (int)blockIdx.x * BM;
  const int oCol0 = (int)blockIdx.y * BN;
  const int rowBase = ty * TM;                // 0..120
  const int colBase = tx * TN;                // 0..60

  float acc[TM][TN];
#pragma unroll
  for (int i = 0; i < TM; ++i)
#pragma unroll
    for (int j = 0; j < TN; ++j)
      acc[i][j] = -INFINITY;

#ifdef USE_TDM
  const int wave = __builtin_amdgcn_readfirstlane(tid >> 5);
  const float* xg = x + (size_t)(bRow0 + wave * XROWS_PER_WAVE) * IN_DIM;
  const float* wg = w + (size_t)(oCol0 + wave * WROWS_PER_WAVE) * IN_DIM;
  const uint32_t xlds0 = (uint32_t)(uintptr_t)&xs[0][wave * XROWS_PER_WAVE * LDS_STRIDE];
  const uint32_t xlds1 = (uint32_t)(uintptr_t)&xs[1][wave * XROWS_PER_WAVE * LDS_STRIDE];
  const uint32_t wlds0 = (uint32_t)(uintptr_t)&ws[0][wave * WROWS_PER_WAVE * LDS_STRIDE];
  const uint32_t wlds1 = (uint32_t)(uintptr_t)&ws[1][wave * WROWS_PER_WAVE * LDS_STRIDE];

  // Prologue: DMA tile 0 into buffer 0.
  tdm_load_2d(xg, xlds0, XROWS_PER_WAVE);
  tdm_load_2d(wg, wlds0, WROWS_PER_WAVE);
#endif

  for (int t = 0; t < KTILES; ++t) {
    const int buf = t & 1;

#ifdef USE_TDM
    if (t + 1 < KTILES) {
      // DMA next tile into the other buffer, overlapping this tile's compute.
      const int k0n = (t + 1) * BK;
      tdm_load_2d(xg + k0n, ((t + 1) & 1) ? xlds1 : xlds0, XROWS_PER_WAVE);
      tdm_load_2d(wg + k0n, ((t + 1) & 1) ? wlds1 : wlds0, WROWS_PER_WAVE);
      // TDM ops from one wave complete in order: <=2 outstanding means the
      // 2 DMAs for tile t have landed in LDS.
      __builtin_amdgcn_s_wait_tensorcnt(2);
    } else {
      __builtin_amdgcn_s_wait_tensorcnt(0);
    }
#else
    // Synchronous fallback: vectorized global loads + LDS stores, same layout.
    const int k0 = t * BK;
#pragma unroll
    for (int q = 0; q < 4; ++q) {                 // 128*32 f32 / 256 thr = 4 x float4
      const int idx = tid + q * THREADS;          // 0..1023
      const int row = idx >> 3;                   // 8 float4 per 32-f32 row
      const int k4  = (idx & 7) << 2;
      const float4 v = *(const float4*)(x + (size_t)(bRow0 + row) * IN_DIM + k0 + k4);
      float* d = &xs[buf][row * LDS_STRIDE + k4];
      d[0] = v.x; d[1] = v.y; d[2] = v.z; d[3] = v.w;
    }
#pragma unroll
    for (int q = 0; q < 2; ++q) {                 // 64*32 f32 / 256 thr = 2 x float4
      const int idx = tid + q * THREADS;          // 0..511
      const int row = idx >> 3;
      const int k4  = (idx & 7) << 2;
      const float4 v = *(const float4*)(w + (size_t)(oCol0 + row) * IN_DIM + k0 + k4);
      float* d = &ws[buf][row * LDS_STRIDE + k4];
      d[0] = v.x; d[1] = v.y; d[2] = v.z; d[3] = v.w;
    }
#endif
    __syncthreads();

    const float* xb = &xs[buf][rowBase * LDS_STRIDE];
    const float* wb = &ws[buf][colBase * LDS_STRIDE];
#pragma unroll 8
    for (int kk = 0; kk < BK; ++kk) {
      float xv[TM], wv[TN];
#pragma unroll
      for (int i = 0; i < TM; ++i) xv[i] = xb[i * LDS_STRIDE + kk];  // wave-broadcast
#pragma unroll
      for (int j = 0; j < TN; ++j) wv[j] = wb[j * LDS_STRIDE + kk];  // conflict-free banks
#pragma unroll
      for (int i = 0; i < TM; ++i)
#pragma unroll
        for (int j = 0; j < TN; ++j)
          acc[i][j] = fmaxf(acc[i][j], xv[i] + wv[j]);
    }
    __syncthreads();
  }

  // Bias fold + store (one float4 per thread row; lanes cover 256B/row).
  float bv[TN];
#pragma unroll
  for (int j = 0; j < TN; ++j) bv[j] = bias[oCol0 + colBase + j];

#pragma unroll
  for (int i = 0; i < TM; ++i) {
    float4 o;
    o.x = fmaxf(acc[i][0], bv[0]);
    o.y = fmaxf(acc[i][1], bv[1]);
    o.z = fmaxf(acc[i][2], bv[2]);
    o.w = fmaxf(acc[i][3], bv[3]);
    *(float4*)(out + (size_t)(bRow0 + rowBase + i) * OUT_DIM + oCol0 + colBase) = o;
  }
}

extern "C" void kernel_launch(void* const* d_in, const int* in_sizes, int n_in,
                              void* d_out, int out_size, void* d_ws, size_t ws_size,
                              hipStream_t stream) {
  (void)in_sizes; (void)n_in; (void)d_ws; (void)ws_size; (void)out_size;
  const float* x    = (const float*)d_in[0];   // [1024, 512]
  const float* wgt  = (const float*)d_in[1];   // [512, 512]
  const float* bias = (const float*)d_in[2];   // [512]
  float* out        = (float*)d_out;           // [1024, 512]

  dim3 grid(B_DIM / BM, OUT_DIM / BN);         // (8, 8) = 64 blocks
  dim3 block(THREADS);                         // 256 threads = 8 wave32
  maxplus_kernel<<<grid, block, 0, stream>>>(x, wgt, bias, out);
}